// AttentionPooling_46059229282478
// MI455X (gfx1250) — compile-verified
//
#include <hip/hip_runtime.h>
#include <math.h>

// ---------------------------------------------------------------------------
// AttentionPooling for MI455X (gfx1250, wave32, WMMA + Tensor Data Mover).
// Shapes: B=8, S=512, H=768, NH=4, DH=192, N=4096 spans (width<=8), FF=3072.
// Strategy:
//   k1a: q = Wq @ dq + bq                                  (tiny)
//   k1b: u[h] = Wk_h^T q_h, qb[h] = q_h . bk_h             (tiny; kills K GEMM)
//   k2 : x = token_reps + sinusoidal PE                    (elementwise)
//   k3 : scores[b,h,s] = (u_h . x[b,s] + qb[h]) / sqrt(DH) (wave reductions)
//   k4 : v = x @ Wv^T + bv            -- WMMA f32 16x16x4 GEMM, 64x64 tiles
//   k5 : span softmax (<=8 keys) + weighted sum of v rows  (VALU, L2-resident v)
//   k6 : attn @ out_w^T + out_b, +dq, LayerNorm  -> h1 (in-place over attn)
//        -- WMMA; weight tiles staged via TDM tensor_load_to_lds (transposed
//           tile with TDM pad -> LDS stride 17, conflict-free frag reads)
//   k7 : FFN fused: relu(h1@W1+b1)@W2+b2, +h1, LayerNorm, *mask -> out
//        -- WMMA; W1/W2 tiles DMA'd by TDM; 3072-wide intermediate in LDS
// Workspace: ~126 MB of d_ws (q/u/qb + x + v + scores + attn(=h1 in place)).
// ---------------------------------------------------------------------------

typedef __attribute__((ext_vector_type(2))) float v2f;
typedef __attribute__((ext_vector_type(8))) float v8f;

constexpr int kB  = 8;
constexpr int kS  = 512;
constexpr int kH  = 768;
constexpr int kNH = 4;
constexpr int kDH = 192;
constexpr int kN  = 4096;
constexpr int kFF = 3072;
constexpr int kBS = kB * kS;    // 4096 rows of x / v
constexpr int kBN = kB * kN;    // 32768 span rows

__device__ __forceinline__ v8f zero8() {
  v8f z = {0.f, 0.f, 0.f, 0.f, 0.f, 0.f, 0.f, 0.f};
  return z;
}

// V_WMMA_F32_16X16X4_F32 : D(16x16,f32) = A(16x4,f32) * B(4x16,f32) + C
__device__ __forceinline__ v8f wmma_f32(v2f a, v2f b, v8f c) {
  return __builtin_amdgcn_wmma_f32_16x16x4_f32(
      /*neg_a=*/false, a, /*neg_b=*/false, b,
      /*c_mod=*/(short)0, c, /*reuse_a=*/false, /*reuse_b=*/false);
}

// A fragment: 16x4 tile. Lanes 0-15: M=lane, K=k0,k0+1; lanes 16-31: K=k0+2,3.
__device__ __forceinline__ v2f frag_a(const float* A, int lda, int k0, int lane) {
  int m  = lane & 15;
  int kk = k0 + ((lane >> 4) << 1);
  v2f a;
  a.x = A[m * lda + kk];
  a.y = A[m * lda + kk + 1];
  return a;
}

// B fragment from [k][n]-major tile.
__device__ __forceinline__ v2f frag_b(const float* Bp, int ldb, int k0, int n0,
                                      int lane) {
  int n  = n0 + (lane & 15);
  int kk = k0 + ((lane >> 4) << 1);
  v2f b;
  b.x = Bp[kk * ldb + n];
  b.y = Bp[(kk + 1) * ldb + n];
  return b;
}

// B fragment from [n][k]-major (transposed) tile: B[k][n] = Bt[n][k].
__device__ __forceinline__ v2f frag_bt(const float* Bt, int ldt, int k0, int n0,
                                       int lane) {
  int n  = n0 + (lane & 15);
  int kk = k0 + ((lane >> 4) << 1);
  v2f b;
  b.x = Bt[n * ldt + kk];
  b.y = Bt[n * ldt + kk + 1];
  return b;
}

// --------------------- Tensor Data Mover (CDNA5 TDM) -----------------------
#if __has_builtin(__builtin_amdgcn_tensor_load_to_lds) && \
    __has_builtin(__builtin_amdgcn_s_wait_tensorcnt)
#define USE_TDM 1
typedef __attribute__((ext_vector_type(4))) unsigned int u32x4;
typedef __attribute__((ext_vector_type(8))) int i32x8;
typedef __attribute__((ext_vector_type(4))) int i32x4;

// Issue one 2D f32 tile load (tile_d0 x tile_d1 elements, global row stride
// `gstride` elements). pad16: insert 1 DWORD of LDS padding after every 16
// DWORDs (-> LDS row stride 17 for 16-wide tiles; bank-conflict-free since
// gcd(17,64)==1). Descriptor packing per CDNA5 ISA.
// This toolchain exposes the 6-arg builtin:
//   (u32x4 g0, i32x8 g1, i32x4 g2, i32x4 g3, i32x8 extra, i32 cpol)
__device__ __forceinline__ void tdm_load_tile_f32(uint32_t lds_byte,
                                                  const float* gsrc,
                                                  uint32_t tile_d0,
                                                  uint32_t tile_d1,
                                                  uint64_t gstride,
                                                  bool pad16) {
  uint64_t ga = (uint64_t)(uintptr_t)gsrc;
  u32x4 g0;
  g0.x = 1u;                                  // count=1, user descriptor
  g0.y = lds_byte;                            // lds_addr[31:0]
  g0.z = (uint32_t)ga;                        // global_addr[31:0]
  g0.w = (uint32_t)((ga >> 32) & 0x1FFFFFFu)  // global_addr[56:32]
         | (2u << 30);                        // type = 2 ("image")
  uint32_t w0 = (2u << 16);                   // data_size = 4 bytes
  if (pad16) w0 |= (1u << 20)                 // pad_enable
               | (3u << 22)                   // pad_interval: 16 DWORDs
               | (0u << 25);                  // pad_amount: 1 DWORD
  // tensor dims == tile dims (tile corner addressing; everything in bounds)
  uint32_t td0 = tile_d0, td1 = tile_d1;
  uint32_t w1 = (td0 & 0xFFFFu) << 16;                          // dim0[15:0]
  uint32_t w2 = ((td0 >> 16) & 0xFFFFu) | ((td1 & 0xFFFFu) << 16);
  uint32_t w3 = ((td1 >> 16) & 0xFFFFu) | ((tile_d0 & 0xFFFFu) << 16);
  uint32_t w4 = (tile_d1 & 0xFFFFu);                            // tile_dim2=0
  uint32_t w5 = (uint32_t)(gstride & 0xFFFFFFFFu);              // d0 stride
  uint32_t w6 = (uint32_t)((gstride >> 32) & 0xFFFFu);          // d1 stride=0
  i32x8 g1 = {(int)w0, (int)w1, (int)w2, (int)w3,
              (int)w4, (int)w5, (int)w6, 0};
  i32x4 zz4 = {0, 0, 0, 0};
  i32x8 zz8 = {0, 0, 0, 0, 0, 0, 0, 0};
  __builtin_amdgcn_tensor_load_to_lds(g0, g1, zz4, zz4, zz8, 0);
}
#endif

// ------------------------- k1a: q projection -------------------------------
__global__ void __launch_bounds__(256)
qproj_kernel(const float* __restrict__ dq, const float* __restrict__ w,
             const float* __restrict__ b_in, float* __restrict__ q) {
  int lane = threadIdx.x & 31, wv = threadIdx.x >> 5;
  int r = blockIdx.x * 8 + wv;            // 0..767
  const float* wr = w + r * kH;           // Wq row r
  float acc = 0.f;
  for (int j = lane; j < kH; j += 32) acc += wr[j] * dq[j];
  for (int off = 16; off; off >>= 1) acc += __shfl_xor(acc, off, 32);
  if (lane == 0) q[r] = acc + b_in[r];
}

// --------------- k1b: folded key vectors u, score bias qb ------------------
__global__ void __launch_bounds__(256)
uvec_kernel(const float* __restrict__ w, const float* __restrict__ b_in,
            const float* __restrict__ q, float* __restrict__ u,
            float* __restrict__ qb) {
  int t = blockIdx.x * 256 + threadIdx.x;  // < NH*H = 3072
  int h = t / kH, j = t % kH;
  float acc = 0.f;
  for (int d = 0; d < kDH; ++d)
    acc += w[(kH + h * kDH + d) * kH + j] * q[h * kDH + d];
  u[t] = acc;
  if (j == 0) {
    float s = 0.f;
    for (int d = 0; d < kDH; ++d)
      s += q[h * kDH + d] * b_in[kH + h * kDH + d];
    qb[h] = s;
  }
}

// ------------------------- k2: x = tokens + PE -----------------------------
__global__ void __launch_bounds__(256)
posenc_kernel(const float* __restrict__ tok, float* __restrict__ x) {
  int idx = blockIdx.x * 256 + threadIdx.x;        // < B*S*H
  int j = idx % kH;
  int s = (idx / kH) & (kS - 1);
  float i2  = (float)(j & ~1);
  float div = expf(-logf(10000.0f) * i2 / (float)kH);
  float ang = (float)s * div;
  float pe  = (j & 1) ? cosf(ang) : sinf(ang);
  x[idx] = tok[idx] + pe;
}

// ------------------------- k3: scores[b,h,s] -------------------------------
__global__ void __launch_bounds__(256)
scores_kernel(const float* __restrict__ x, const float* __restrict__ u,
              const float* __restrict__ qb, float* __restrict__ scores) {
  int lane = threadIdx.x & 31, wv = threadIdx.x >> 5;
  int gid = blockIdx.x * 8 + wv;           // over B*NH*S = 16384
  int s = gid & (kS - 1);
  int h = (gid >> 9) & (kNH - 1);
  int b = gid >> 11;
  const float* xr = x + (b * kS + s) * kH;
  const float* ur = u + h * kH;
  float acc = 0.f;
  for (int j = lane; j < kH; j += 32) acc += xr[j] * ur[j];
  for (int off = 16; off; off >>= 1) acc += __shfl_xor(acc, off, 32);
  if (lane == 0) scores[gid] = (acc + qb[h]) * 0.0721687836f;  // 1/sqrt(192)
}

// ------------------- k4: v = x @ Wv^T + bv (WMMA GEMM) ---------------------
__global__ void __launch_bounds__(256)
v_gemm_kernel(const float* __restrict__ x, const float* __restrict__ w,
              const float* __restrict__ b_in, float* __restrict__ v) {
  const int tid = threadIdx.x, lane = tid & 31, wave = tid >> 5;
  const int m0 = blockIdx.y * 64;
  const int n0b = blockIdx.x * 64;
  extern __shared__ float sm[];
  float (*As)[17] = (float (*)[17])sm;             // 64 x 17
  float (*Bs)[65] = (float (*)[65])(sm + 64 * 17); // 16 x 65
  v8f acc[2] = {zero8(), zero8()};
  for (int kt = 0; kt < kH; kt += 16) {
    __syncthreads();
    for (int i = tid; i < 64 * 16; i += 256) {     // A: 64x16, coalesced in k
      int r = i >> 4, c = i & 15;
      As[r][c] = x[(m0 + r) * kH + kt + c];
    }
    for (int i = tid; i < 16 * 64; i += 256) {     // B[k][n] = Wv[n][k]
      int k = i & 15, n = i >> 4;                  // k fastest -> coalesced
      Bs[k][n] = w[(2 * kH + n0b + n) * kH + kt + k];
    }
    __syncthreads();
#pragma unroll
    for (int t = 0; t < 2; ++t) {
      int tt = wave * 2 + t;
      int ti = tt >> 2, tj = tt & 3;
#pragma unroll
      for (int ks = 0; ks < 16; ks += 4) {
        v2f a = frag_a(&As[ti * 16][0], 17, ks, lane);
        v2f b = frag_b(&Bs[0][0], 65, ks, tj * 16, lane);
        acc[t] = wmma_f32(a, b, acc[t]);
      }
    }
  }
#pragma unroll
  for (int t = 0; t < 2; ++t) {
    int tt = wave * 2 + t;
    int ti = tt >> 2, tj = tt & 3;
    int n  = n0b + tj * 16 + (lane & 15);
    int mb = m0 + ti * 16 + ((lane >> 4) << 3);
    float bb = b_in[2 * kH + n];
#pragma unroll
    for (int r = 0; r < 8; ++r) v[(mb + r) * kH + n] = acc[t][r] + bb;
  }
}

// --------------------- k5: span softmax + gather ---------------------------
__global__ void __launch_bounds__(128)
span_attn_kernel(const float* __restrict__ scores, const float* __restrict__ v,
                 const int* __restrict__ span_ids,
                 const int* __restrict__ span_masks, float* __restrict__ attn) {
  int row = blockIdx.x;                    // b*N + n
  int b = row >> 12;                       // N = 4096
  int tid = threadIdx.x;
  __shared__ float wgt[kNH][8];
  __shared__ int sh_st, sh_w, sh_mk;
  if (tid == 0) {
    int st = span_ids[row * 2], en = span_ids[row * 2 + 1];
    sh_st = st; sh_w = en - st; sh_mk = span_masks[row];
  }
  __syncthreads();
  int st = sh_st, wdt = sh_w, mk = sh_mk;
  if (mk == 0) {                           // final output is zeroed anyway
    for (int j = tid; j < kH; j += 128) attn[row * kH + j] = 0.f;
    return;
  }
  if (tid < kNH) {
    const float* sc = scores + (b * kNH + tid) * kS + st;
    float m = -1e30f;
    for (int p = 0; p < wdt; ++p) m = fmaxf(m, sc[p]);
    float e[8];
    float ssum = 0.f;
    for (int p = 0; p < wdt; ++p) { e[p] = expf(sc[p] - m); ssum += e[p]; }
    float inv = 1.f / ssum;
    for (int p = 0; p < 8; ++p) wgt[tid][p] = (p < wdt) ? e[p] * inv : 0.f;
  }
  __syncthreads();
  for (int j = tid; j < kH; j += 128) {
    int h = j / kDH;
    const float* vp = v + ((b << 9) + st) * kH + j;   // S = 512
    float acc = 0.f;
    for (int p = 0; p < wdt; ++p) acc += wgt[h][p] * vp[p * kH];
    attn[row * kH + j] = acc;
  }
}

// ------ k6: h1 = LN(attn @ out_w^T + out_b + dq)  (in-place over attn) -----
// 16 rows x full 768 cols per block; 8 waves * 6 C-tiles. B tile = out_w^T
// staged by TDM as a natural-order [n][k] tile with 1-DWORD/16-DWORD padding
// (LDS stride 17). Epilogue Y spills into the A buffer.
__global__ void __launch_bounds__(256)
outproj_ln_kernel(float* __restrict__ attn, const float* __restrict__ out_w,
                  const float* __restrict__ out_b, const float* __restrict__ dq,
                  const float* __restrict__ ln_g, const float* __restrict__ ln_b) {
  const int tid = threadIdx.x, lane = tid & 31, wave = tid >> 5;
  const int m0 = blockIdx.x * 16;
  extern __shared__ float sm[];
  float (*As)[kH + 1] = (float (*)[kH + 1])sm;            // 16 x 769 (A; -> Y)
  float (*Bt)[17] = (float (*)[17])(sm + 16 * (kH + 1));  // 768 x 17 (B^T)
  __shared__ float red1[16][17], red2[16][17], mu[16], rsig[16];
  for (int r = 0; r < 16; ++r)
    for (int c = tid; c < kH; c += 256) As[r][c] = attn[(m0 + r) * kH + c];
  v8f acc[6];
#pragma unroll
  for (int t = 0; t < 6; ++t) acc[t] = zero8();
  for (int kt = 0; kt < kH; kt += 16) {
    __syncthreads();
#if defined(USE_TDM)
    if (wave == 0) {   // DMA out_w[n][kt..kt+15] tile: 16 x 768 rows
      tdm_load_tile_f32((uint32_t)(uintptr_t)&Bt[0][0], out_w + kt,
                        /*tile_d0=*/16, /*tile_d1=*/kH, /*gstride=*/kH,
                        /*pad16=*/true);
      __builtin_amdgcn_s_wait_tensorcnt(0);
    }
#else
    for (int i = tid; i < 16 * kH; i += 256) {
      int k = i & 15, n = i >> 4;               // k fastest -> coalesced
      Bt[n][k] = out_w[n * kH + kt + k];
    }
#endif
    __syncthreads();
#pragma unroll
    for (int ks = 0; ks < 16; ks += 4) {
      v2f a = frag_a(&As[0][0], kH + 1, kt + ks, lane);
#pragma unroll
      for (int t = 0; t < 6; ++t) {
        v2f b = frag_bt(&Bt[0][0], 17, ks, (wave * 6 + t) * 16, lane);
        acc[t] = wmma_f32(a, b, acc[t]);
      }
    }
  }
  __syncthreads();
#pragma unroll
  for (int t = 0; t < 6; ++t) {                 // spill C tiles -> Y (in As)
    int n  = (wave * 6 + t) * 16 + (lane & 15);
    int mb = (lane >> 4) << 3;
#pragma unroll
    for (int r = 0; r < 8; ++r) As[mb + r][n] = acc[t][r];
  }
  __syncthreads();
  for (int r = 0; r < 16; ++r)
    for (int c = tid; c < kH; c += 256) As[r][c] += out_b[c] + dq[c];
  __syncthreads();
  {                                             // LayerNorm partials
    int r = tid >> 4, c0 = tid & 15;
    float s1 = 0.f, s2 = 0.f;
    for (int c = c0; c < kH; c += 16) { float q = As[r][c]; s1 += q; s2 += q * q; }
    red1[r][c0] = s1; red2[r][c0] = s2;
  }
  __syncthreads();
  if ((tid & 15) == 0) {
    int r = tid >> 4;
    float s1 = 0.f, s2 = 0.f;
    for (int i = 0; i < 16; ++i) { s1 += red1[r][i]; s2 += red2[r][i]; }
    float m = s1 / (float)kH;
    mu[r] = m;
    rsig[r] = rsqrtf(s2 / (float)kH - m * m + 1e-5f);
  }
  __syncthreads();
  for (int r = 0; r < 16; ++r)
    for (int c = tid; c < kH; c += 256)
      attn[(m0 + r) * kH + c] = (As[r][c] - mu[r]) * rsig[r] * ln_g[c] + ln_b[c];
}

// ------ k7: out = LN(relu(h1@W1+b1)@W2 + b2 + h1) * mask  (fully fused) ----
// 16 rows/block; 3072 intermediate as 4 chunks of 768 in LDS. W1/W2 tiles
// (16 x 768, naturally contiguous) DMA'd by TDM into a stride-768 LDS tile.
__global__ void __launch_bounds__(256)
ffn_kernel(const float* __restrict__ h1, const float* __restrict__ w1,
           const float* __restrict__ b1, const float* __restrict__ w2,
           const float* __restrict__ b2, const float* __restrict__ ln_g,
           const float* __restrict__ ln_b, const int* __restrict__ span_masks,
           float* __restrict__ out) {
  const int tid = threadIdx.x, lane = tid & 31, wave = tid >> 5;
  const int m0 = blockIdx.x * 16;
  extern __shared__ float sm[];
  float (*Hs)[kH + 1]  = (float (*)[kH + 1])sm;                     // h1 tile
  float (*Act)[kH + 1] = (float (*)[kH + 1])(sm + 16 * (kH + 1));   // a chunk
  float* Bsf = sm + 2 * 16 * (kH + 1);           // 16 x 768 flat [k][n] / Y
  __shared__ float red1[16][17], red2[16][17], mu[16], rsig[16];
  for (int r = 0; r < 16; ++r)
    for (int c = tid; c < kH; c += 256) Hs[r][c] = h1[(m0 + r) * kH + c];
  v8f acc2[6];
#pragma unroll
  for (int t = 0; t < 6; ++t) acc2[t] = zero8();
  for (int ch = 0; ch < 4; ++ch) {
    v8f acc1[6];
#pragma unroll
    for (int t = 0; t < 6; ++t) acc1[t] = zero8();
    for (int kt = 0; kt < kH; kt += 16) {       // GEMM1: h1 @ W1[:,chunk]
      __syncthreads();
#if defined(USE_TDM)
      if (wave == 0) {
        tdm_load_tile_f32((uint32_t)(uintptr_t)Bsf,
                          w1 + (size_t)kt * kFF + ch * kH,
                          /*tile_d0=*/kH, /*tile_d1=*/16, /*gstride=*/kFF,
                          /*pad16=*/false);
        __builtin_amdgcn_s_wait_tensorcnt(0);
      }
#else
      for (int k = 0; k < 16; ++k)
        for (int n = tid; n < kH; n += 256)     // n fastest -> coalesced
          Bsf[k * kH + n] = w1[(kt + k) * kFF + ch * kH + n];
#endif
      __syncthreads();
#pragma unroll
      for (int ks = 0; ks < 16; ks += 4) {
        v2f a = frag_a(&Hs[0][0], kH + 1, kt + ks, lane);
#pragma unroll
        for (int t = 0; t < 6; ++t) {
          v2f b = frag_b(Bsf, kH, ks, (wave * 6 + t) * 16, lane);
          acc1[t] = wmma_f32(a, b, acc1[t]);
        }
      }
    }
    __syncthreads();
#pragma unroll
    for (int t = 0; t < 6; ++t) {               // Act = relu(acc1 + b1)
      int n  = (wave * 6 + t) * 16 + (lane & 15);
      int mb = (lane >> 4) << 3;
      float bb = b1[ch * kH + n];
#pragma unroll
      for (int r = 0; r < 8; ++r) {
        float q = acc1[t][r] + bb;
        Act[mb + r][n] = q > 0.f ? q : 0.f;
      }
    }
    for (int kt = 0; kt < kH; kt += 16) {       // GEMM2: Act @ W2[chunk,:]
      __syncthreads();
#if defined(USE_TDM)
      if (wave == 0) {
        tdm_load_tile_f32((uint32_t)(uintptr_t)Bsf,
                          w2 + (size_t)(ch * kH + kt) * kH,
                          /*tile_d0=*/kH, /*tile_d1=*/16, /*gstride=*/kH,
                          /*pad16=*/false);
        __builtin_amdgcn_s_wait_tensorcnt(0);
      }
#else
      for (int k = 0; k < 16; ++k)
        for (int n = tid; n < kH; n += 256)
          Bsf[k * kH + n] = w2[(ch * kH + kt + k) * kH + n];
#endif
      __syncthreads();
#pragma unroll
      for (int ks = 0; ks < 16; ks += 4) {
        v2f a = frag_a(&Act[0][0], kH + 1, kt + ks, lane);
#pragma unroll
        for (int t = 0; t < 6; ++t) {
          v2f b = frag_b(Bsf, kH, ks, (wave * 6 + t) * 16, lane);
          acc2[t] = wmma_f32(a, b, acc2[t]);
        }
      }
    }
  }
  __syncthreads();
#pragma unroll
  for (int t = 0; t < 6; ++t) {                 // spill -> Y (in Bsf)
    int n  = (wave * 6 + t) * 16 + (lane & 15);
    int mb = (lane >> 4) << 3;
#pragma unroll
    for (int r = 0; r < 8; ++r) Bsf[(mb + r) * kH + n] = acc2[t][r];
  }
  __syncthreads();
  for (int r = 0; r < 16; ++r)                  // +b2 + residual h1
    for (int c = tid; c < kH; c += 256) Bsf[r * kH + c] += b2[c] + Hs[r][c];
  __syncthreads();
  {
    int r = tid >> 4, c0 = tid & 15;
    float s1 = 0.f, s2 = 0.f;
    for (int c = c0; c < kH; c += 16) {
      float q = Bsf[r * kH + c];
      s1 += q; s2 += q * q;
    }
    red1[r][c0] = s1; red2[r][c0] = s2;
  }
  __syncthreads();
  if ((tid & 15) == 0) {
    int r = tid >> 4;
    float s1 = 0.f, s2 = 0.f;
    for (int i = 0; i < 16; ++i) { s1 += red1[r][i]; s2 += red2[r][i]; }
    float m = s1 / (float)kH;
    mu[r] = m;
    rsig[r] = rsqrtf(s2 / (float)kH - m * m + 1e-5f);
  }
  __syncthreads();
  for (int r = 0; r < 16; ++r) {
    int row = m0 + r;
    float mk = (span_masks[row] != 0) ? 1.f : 0.f;
    for (int c = tid; c < kH; c += 256)
      out[row * kH + c] =
          ((Bsf[r * kH + c] - mu[r]) * rsig[r] * ln_g[c] + ln_b[c]) * mk;
  }
}

// ---------------------------------------------------------------------------
extern "C" void kernel_launch(void* const* d_in, const int* in_sizes, int n_in,
                              void* d_out, int out_size, void* d_ws,
                              size_t ws_size, hipStream_t stream) {
  (void)in_sizes; (void)n_in; (void)out_size; (void)ws_size;
  const float* tok  = (const float*)d_in[0];   // [B,S,H]
  const float* dq   = (const float*)d_in[1];   // [H]
  const float* ipw  = (const float*)d_in[2];   // [3H,H]
  const float* ipb  = (const float*)d_in[3];   // [3H]
  const float* outw = (const float*)d_in[4];   // [H,H]
  const float* outb = (const float*)d_in[5];   // [H]
  const float* lng  = (const float*)d_in[6];   // [H]
  const float* lnb  = (const float*)d_in[7];   // [H]
  const float* w1   = (const float*)d_in[8];   // [H,4H]
  const float* b1   = (const float*)d_in[9];   // [4H]
  const float* w2   = (const float*)d_in[10];  // [4H,H]
  const float* b2   = (const float*)d_in[11];  // [H]
  const int* sids   = (const int*)d_in[12];    // [B,N,2]
  const int* smask  = (const int*)d_in[13];    // [B,N]
  float* out = (float*)d_out;                  // [B,N,H]

  float* ws = (float*)d_ws;                    // needs ~126 MB
  float* q      = ws;                          // 768
  float* u      = ws + 768;                    // 4*768
  float* qb     = ws + 768 + 4 * 768;          // 4
  float* x      = ws + 4096;                   // BS*H
  float* v      = x + (size_t)kBS * kH;        // BS*H
  float* scores = v + (size_t)kBS * kH;        // B*NH*S
  float* attn   = scores + kB * kNH * kS;      // BN*H (becomes h1 in place)

  qproj_kernel<<<kH / 8, 256, 0, stream>>>(dq, ipw, ipb, q);
  uvec_kernel<<<(kNH * kH) / 256, 256, 0, stream>>>(ipw, ipb, q, u, qb);
  posenc_kernel<<<(kBS * kH) / 256, 256, 0, stream>>>(tok, x);
  scores_kernel<<<(kB * kNH * kS) / 8, 256, 0, stream>>>(x, u, qb, scores);

  size_t lds_v = (64 * 17 + 16 * 65) * sizeof(float);
  v_gemm_kernel<<<dim3(kH / 64, kBS / 64), 256, lds_v, stream>>>(x, ipw, ipb, v);

  span_attn_kernel<<<kBN, 128, 0, stream>>>(scores, v, sids, smask, attn);

  size_t lds6 = (size_t)(16 * (kH + 1) + kH * 17) * sizeof(float);  // ~99 KB
  outproj_ln_kernel<<<kBN / 16, 256, lds6, stream>>>(attn, outw, outb, dq,
                                                     lng, lnb);

  size_t lds7 = (size_t)(2 * 16 * (kH + 1) + 16 * kH) * sizeof(float); // ~144 KB
  ffn_kernel<<<kBN / 16, 256, lds7, stream>>>(attn, w1, b1, w2, b2, lng, lnb,
                                              smask, out);
}